// SphereByCurvature_86629490360623
// MI455X (gfx1250) — compile-verified
//
#include <hip/hip_runtime.h>
#include <cstdint>

#define NRAYS_PER_BLOCK 256
#define N_ITER 6
#define DAMPING 0.95f
#define EPS_F 1e-12f
#define VALID_TOL 1e-4f

// fast hardware approximations (v_rcp_f32 / v_rsq_f32)
__device__ __forceinline__ float rcp_f(float x) { return __builtin_amdgcn_rcpf(x); }
__device__ __forceinline__ float rsq_f(float x) { return __builtin_amdgcn_rsqf(x); }

// GNU-vector int4 with explicit address spaces, matching the builtin's
// expected parameter types exactly:
//   __builtin_amdgcn_global_load_async_to_lds_b128(v4i AS1*, v4i AS3*, Ii, Ii)
typedef int v4i __attribute__((vector_size(16)));
typedef __attribute__((address_space(1))) v4i gv4i;   // global
typedef __attribute__((address_space(3))) v4i lv4i;   // LDS

// Parameter block layout in d_ws (floats)
#define P_R      0   // 9 floats, row-major R = tf[:3,:3]
#define P_TV     9   // 3 floats, tv = tf[:3,3] + tf[:3,0] * (-a0*extent)
#define P_CE2    12  // C_eff^2
#define P_INVS2  13  // 1/scale^2
#define P_SCE    14  // scale * C_eff
#define P_KCE    15  // C_eff / scale
#define P_RMAX2  16  // (diameter/2)^2
#define NPARAMS  17

// ---------------------------------------------------------------------------
// Tiny scalar setup: surface_tf / next_tf + folded parameter block.
// ---------------------------------------------------------------------------
__global__ void sphere_setup_kernel(const float* __restrict__ tf,
                                    const float* __restrict__ diameter,
                                    const float* __restrict__ C,
                                    const float* __restrict__ anchors,
                                    const float* __restrict__ scale,
                                    const unsigned char* __restrict__ normalize,
                                    float* __restrict__ out_tf,   // d_out + 5N (32 floats)
                                    float* __restrict__ params)   // d_ws
{
    if (threadIdx.x != 0 || blockIdx.x != 0) return;

    const float d  = diameter[0];
    const float c0 = C[0];
    const float a0 = anchors[0];
    const float a1 = anchors[1];
    const float sc = scale[0];
    const bool  nrm = (normalize[0] != 0);   // first byte works for bool8 or int32 0/1

    const float Ce  = nrm ? (c0 * 2.0f / d) : c0;
    const float s2  = sc * sc;
    const float r2m = (0.5f * d) * (0.5f * d);

    // extent = sag((d/2)^2), full precision (runs once)
    const float u    = r2m / s2;
    const float arg  = fmaxf(1.0f - Ce * Ce * u, EPS_F);
    const float root = sqrtf(arg);
    const float extent = sc * Ce * u / (1.0f + root);

    const float dx0 = -a0 * extent;
    const float dx1 = (a1 - a0) * extent;

    // surface_tf = tf @ Tx(dx0); next_tf = tf @ Tx(dx1)
    // right-mul by Tx only changes column 3: col3' = tf[:,0]*dx + tf[:,3]
    for (int r = 0; r < 4; ++r) {
        for (int cc = 0; cc < 4; ++cc) {
            const float v = tf[r * 4 + cc];
            const float col0 = tf[r * 4 + 0];
            out_tf[r * 4 + cc]      = (cc == 3) ? fmaf(col0, dx0, v) : v;
            out_tf[16 + r * 4 + cc] = (cc == 3) ? fmaf(col0, dx1, v) : v;
        }
    }

    for (int i = 0; i < 3; ++i) {
        for (int j = 0; j < 3; ++j) params[P_R + i * 3 + j] = tf[i * 4 + j];
        params[P_TV + i] = fmaf(tf[i * 4 + 0], dx0, tf[i * 4 + 3]);
    }
    params[P_CE2]   = Ce * Ce;
    params[P_INVS2] = 1.0f / s2;
    params[P_SCE]   = sc * Ce;
    params[P_KCE]   = Ce / sc;
    params[P_RMAX2] = r2m;
}

// ---------------------------------------------------------------------------
// Main ray-trace kernel: async global->LDS staging + damped Newton.
// ---------------------------------------------------------------------------
__global__ __launch_bounds__(NRAYS_PER_BLOCK)
void sphere_trace_kernel(const float* __restrict__ P,
                         const float* __restrict__ V,
                         const float* __restrict__ params,
                         float* __restrict__ out,
                         int n)
{
    __shared__ float sP[3 * NRAYS_PER_BLOCK];   // 3 KB
    __shared__ float sV[3 * NRAYS_PER_BLOCK];   // 3 KB
    __shared__ float sPar[NPARAMS];

    const int tid  = threadIdx.x;
    const int base = blockIdx.x * NRAYS_PER_BLOCK;

    if (tid < NPARAMS) sPar[tid] = params[tid];

    const bool full = (base + NRAYS_PER_BLOCK) <= n;
    if (full) {
        // 3*256 dwords per tile = 192 lanes x one b128 each, per array.
#if __has_builtin(__builtin_amdgcn_global_load_async_to_lds_b128)
        if (tid < 192) {
            const float* gP = P + (size_t)base * 3 + (size_t)tid * 4;
            const float* gV = V + (size_t)base * 3 + (size_t)tid * 4;
            __builtin_amdgcn_global_load_async_to_lds_b128(
                (gv4i*)gP, (lv4i*)(&sP[tid * 4]), 0, 0);
            __builtin_amdgcn_global_load_async_to_lds_b128(
                (gv4i*)gV, (lv4i*)(&sV[tid * 4]), 0, 0);
        }
#if __has_builtin(__builtin_amdgcn_s_wait_asynccnt)
        __builtin_amdgcn_s_wait_asynccnt(0);
#else
        asm volatile("s_wait_asynccnt 0" ::: "memory");
#endif
#else
        // Fallback: coalesced b128 load + LDS store (still one vmem op / lane)
        if (tid < 192) {
            const float4 p4 = *(const float4*)(P + (size_t)base * 3 + (size_t)tid * 4);
            const float4 v4 = *(const float4*)(V + (size_t)base * 3 + (size_t)tid * 4);
            *(float4*)(&sP[tid * 4]) = p4;
            *(float4*)(&sV[tid * 4]) = v4;
        }
#endif
    }
    __syncthreads();

    const int i = base + tid;
    if (i >= n) return;

    float px, py, pz, vx, vy, vz;
    if (full) {
        // stride-3 LDS reads: gcd(3,64)=1 -> bank-conflict free across a wave32
        px = sP[tid * 3 + 0]; py = sP[tid * 3 + 1]; pz = sP[tid * 3 + 2];
        vx = sV[tid * 3 + 0]; vy = sV[tid * 3 + 1]; vz = sV[tid * 3 + 2];
    } else {
        px = P[(size_t)i * 3 + 0]; py = P[(size_t)i * 3 + 1]; pz = P[(size_t)i * 3 + 2];
        vx = V[(size_t)i * 3 + 0]; vy = V[(size_t)i * 3 + 1]; vz = V[(size_t)i * 3 + 2];
    }

    // Parameters
    const float R00 = sPar[P_R + 0], R01 = sPar[P_R + 1], R02 = sPar[P_R + 2];
    const float R10 = sPar[P_R + 3], R11 = sPar[P_R + 4], R12 = sPar[P_R + 5];
    const float R20 = sPar[P_R + 6], R21 = sPar[P_R + 7], R22 = sPar[P_R + 8];
    const float tvx = sPar[P_TV + 0], tvy = sPar[P_TV + 1], tvz = sPar[P_TV + 2];
    const float Ce2   = sPar[P_CE2];
    const float invS2 = sPar[P_INVS2];
    const float sCe   = sPar[P_SCE];
    const float kCe   = sPar[P_KCE];
    const float rmax2 = sPar[P_RMAX2];

    // Pl = (P - tv) @ R   (Pl_j = sum_i d_i * R[i][j]);  Vl = V @ R
    const float dx = px - tvx, dy = py - tvy, dz = pz - tvz;
    const float Plx = fmaf(dx, R00, fmaf(dy, R10, dz * R20));
    const float Ply = fmaf(dx, R01, fmaf(dy, R11, dz * R21));
    const float Plz = fmaf(dx, R02, fmaf(dy, R12, dz * R22));
    const float Vlx = fmaf(vx, R00, fmaf(vy, R10, vz * R20));
    const float Vly = fmaf(vx, R01, fmaf(vy, R11, vz * R21));
    const float Vlz = fmaf(vx, R02, fmaf(vy, R12, vz * R22));

    const float safe_vx = (fabsf(Vlx) > EPS_F) ? Vlx : EPS_F;
    float t = -Plx * rcp_f(safe_vx);

    // 6 damped Newton iterations
#pragma unroll
    for (int it = 0; it < N_ITER; ++it) {
        const float ppx = fmaf(t, Vlx, Plx);
        const float ppy = fmaf(t, Vly, Ply);
        const float ppz = fmaf(t, Vlz, Plz);
        const float r2  = fmaf(ppy, ppy, ppz * ppz);
        const float u   = r2 * invS2;
        const float arg = fmaxf(fmaf(-Ce2, u, 1.0f), EPS_F);
        const float rs  = rsq_f(arg);          // 1/sqrt(arg)
        const float root = arg * rs;           // sqrt(arg)
        const float F   = fmaf(sCe * u, rcp_f(1.0f + root), -ppx);
        const float k   = kCe * rs;            // Ce/(scale*root)
        float dF = fmaf(k, fmaf(ppy, Vly, ppz * Vlz), -Vlx);
        dF = (fabsf(dF) > EPS_F) ? dF : EPS_F;
        t = fmaf(-DAMPING * F, rcp_f(dF), t);
    }

    // Final evaluation: residual, normal, validity
    const float ppx = fmaf(t, Vlx, Plx);
    const float ppy = fmaf(t, Vly, Ply);
    const float ppz = fmaf(t, Vlz, Plz);
    const float r2  = fmaf(ppy, ppy, ppz * ppz);
    const float u   = r2 * invS2;
    const float arg = fmaxf(fmaf(-Ce2, u, 1.0f), EPS_F);
    const float rs  = rsq_f(arg);
    const float root = arg * rs;
    const float F   = fmaf(sCe * u, rcp_f(1.0f + root), -ppx);
    const float k   = kCe * rs;
    const float gy  = k * ppy;
    const float gz  = k * ppz;

    const float invn = rsq_f(fmaf(gy, gy, fmaf(gz, gz, 1.0f)));
    const float nlx = -invn;
    const float nly = gy * invn;
    const float nlz = gz * invn;

    // normals = n_local @ R^T  (out_j = sum_i nl_i * R[j][i])
    const float nx = fmaf(nlx, R00, fmaf(nly, R01, nlz * R02));
    const float ny = fmaf(nlx, R10, fmaf(nly, R11, nlz * R12));
    const float nz = fmaf(nlx, R20, fmaf(nly, R21, nlz * R22));

    const bool valid = (r2 <= rmax2) && (fabsf(F) < VALID_TOL);

    // Output layout: [t:N][normals:3N][valid:N][surface_tf:16][next_tf:16]
    out[i] = t;
    float* nb = out + (size_t)n;
    nb[(size_t)i * 3 + 0] = nx;
    nb[(size_t)i * 3 + 1] = ny;
    nb[(size_t)i * 3 + 2] = nz;
    out[(size_t)4 * n + i] = valid ? 1.0f : 0.0f;
}

// ---------------------------------------------------------------------------
extern "C" void kernel_launch(void* const* d_in, const int* in_sizes, int n_in,
                              void* d_out, int out_size, void* d_ws, size_t ws_size,
                              hipStream_t stream)
{
    const float* P        = (const float*)d_in[0];   // N*3
    const float* V        = (const float*)d_in[1];   // N*3
    const float* tf       = (const float*)d_in[2];   // 16
    const float* diameter = (const float*)d_in[3];   // 1
    const float* C        = (const float*)d_in[4];   // 1
    const float* anchors  = (const float*)d_in[5];   // 2
    const float* scale    = (const float*)d_in[6];   // 1
    const unsigned char* normalize = (const unsigned char*)d_in[7]; // 1 (bool)

    const int n = in_sizes[0] / 3;
    float* out = (float*)d_out;
    float* params = (float*)d_ws;

    hipLaunchKernelGGL(sphere_setup_kernel, dim3(1), dim3(1), 0, stream,
                       tf, diameter, C, anchors, scale, normalize,
                       out + (size_t)5 * n, params);

    const int blocks = (n + NRAYS_PER_BLOCK - 1) / NRAYS_PER_BLOCK;
    hipLaunchKernelGGL(sphere_trace_kernel, dim3(blocks), dim3(NRAYS_PER_BLOCK), 0, stream,
                       P, V, params, out, n);
}